// AnchorToAnchor_27178553049412
// MI455X (gfx1250) — compile-verified
//
#include <hip/hip_runtime.h>

typedef __attribute__((ext_vector_type(2))) float v2f;
typedef __attribute__((ext_vector_type(8))) float v8f;

// ---------------------------------------------------------------------------
// Fixed problem dims (from setup_inputs):
//   N=8, C=256, H=W=64, A=9, bs=8, F=8, K=64, M=A*N*K=4608
// ---------------------------------------------------------------------------

// Stage 1: strided conv as GEMM via V_WMMA_F32_16X16X4_F32.
// Per image n: reg_tile(64 x 16pad) = P(64 x 16384) * W^T(16384 x 16pad).
// Grid: 2048 blocks of 1 wave: bid = kc*32 + n*4 + mt.
// Each wave covers a K-chunk of 256 (64 wmma steps of K=4) and writes its
// partial 16x16 f32 tile to ws (unique slot -> deterministic, no atomics).
__global__ __launch_bounds__(32) void conv_wmma_kernel(
    const float* __restrict__ fb,    // (8,256,64,64)
    const float* __restrict__ wr,    // (9,256,8,8) flat: a*16384 + k
    float* __restrict__ part)        // (64 kc, 8 n, 4 mt, 256)
{
  const int bid  = blockIdx.x;
  const int kc   = bid >> 5;          // 0..63
  const int rem  = bid & 31;
  const int n    = rem >> 2;          // 0..7
  const int mt   = rem & 3;           // 0..3

  const int lane = threadIdx.x;       // 0..31 (wave32)
  const int l15  = lane & 15;
  const int half = lane >> 4;         // 0/1 -> K offset 0/2

  // A fragment row: patch m = mt*16 + l15 -> (fy, fx) block
  const int m   = mt * 16 + l15;
  const int fy8 = (m >> 3) << 3;      // fy*8
  const int fx8 = (m & 7) << 3;       // fx*8
  const float* fbn = fb + (((size_t)n * 256) << 12);   // n*256*4096

  // B fragment col: anchor a = l15 (cols 9..15 zeroed)
  const int  arow  = l15;
  const bool avalid = (arow < 9);
  const float* wra = wr + (arow < 9 ? arow : 0) * 16384;

  v8f acc = {};
  const int kbase = kc * 256 + half * 2;

  #pragma unroll 4
  for (int kk = 0; kk < 64; ++kk) {
    const int kglob = kbase + kk * 4;          // even -> 8B aligned loads
    // A: patch element k -> (c, ky, kx), kx even so (kx,kx+1) same row
    const int c   = kglob >> 6;
    const int r6  = kglob & 63;
    const int ky  = r6 >> 3;
    const int kx  = r6 & 7;
    v2f afrag = *(const v2f*)(fbn + ((c << 12) + ((fy8 + ky) << 6) + fx8 + kx));
    // B: W[k, a]
    v2f bfrag = *(const v2f*)(wra + kglob);
    if (!avalid) { bfrag.x = 0.f; bfrag.y = 0.f; }

    // D = A(16x4) * B(4x16) + C   (full fp32 — conv is bandwidth-bound,
    // so there is zero upside to dropping precision here)
    acc = __builtin_amdgcn_wmma_f32_16x16x4_f32(
        /*neg_a=*/false, afrag, /*neg_b=*/false, bfrag,
        /*c_mod=*/(short)0, acc, /*reuse_a=*/false, /*reuse_b=*/false);
  }

  // D tile: VGPR v, lane l -> M = v + 8*half, N = l15
  float* pp = part + ((((kc << 3) + n) << 2) + mt) * 256;
  const int row0 = half << 3;
  #pragma unroll
  for (int v = 0; v < 8; ++v)
    pp[(row0 + v) * 16 + l15] = acc[v];
}

// Stage 1b: reduce 64 K-chunk partials, add bias, emit offs = 0.8*tanh(reg).
// offs layout: [n][m=64][a=16pad]
__global__ __launch_bounds__(256) void reduce_reg_kernel(
    const float* __restrict__ part, const float* __restrict__ breg,
    float* __restrict__ offs)
{
  const int t = blockIdx.x * 256 + threadIdx.x;   // 0..8191
  const int a = t & 15;
  const int m = (t >> 4) & 63;
  const int n = t >> 10;
  float v = 0.f;
  if (a < 9) {
    const int inner = (m & 15) * 16 + a;
    const int mt    = m >> 4;
    #pragma unroll 4
    for (int kc = 0; kc < 64; ++kc)
      v += part[((((kc << 3) + n) << 2) + mt) * 256 + inner];
    v = 0.8f * tanhf(v + breg[a]);                // ALPHA * bs = 0.8
  }
  offs[t] = v;
}

// Stage 2: bilinear sampling -> rois_feature_b in flat (A,N,K,C) layout.
// One block per (a,n,k); thread = channel.
__global__ __launch_bounds__(256) void sample_kernel(
    const float* __restrict__ fb, const float* __restrict__ offs,
    float* __restrict__ rfb)
{
  const int bid = blockIdx.x;            // 9*8*64 = 4608
  const int a   = bid / 512;
  const int rem = bid & 511;
  const int n   = rem >> 6;
  const int k   = rem & 63;
  const int fy  = k >> 3, fx = k & 7;

  const float off = offs[((n << 6) + k) * 16 + a];
  const float cx = 3.5f + 8.f * (float)fx + off;
  const float cy = 3.5f + 8.f * (float)fy + off;
  const float x0f = floorf(cx), y0f = floorf(cy);
  const float wx = cx - x0f, wy = cy - y0f;
  int x0 = (int)x0f; x0 = x0 < 0 ? 0 : (x0 > 63 ? 63 : x0);
  int y0 = (int)y0f; y0 = y0 < 0 ? 0 : (y0 > 63 ? 63 : y0);
  const int x1 = x0 + 1 > 63 ? 63 : x0 + 1;
  const int y1 = y0 + 1 > 63 ? 63 : y0 + 1;

  const int c = threadIdx.x;
  const float* base = fb + ((((size_t)n << 8) + c) << 12);
  const float v00 = base[(y0 << 6) + x0];
  const float v01 = base[(y0 << 6) + x1];
  const float v10 = base[(y1 << 6) + x0];
  const float v11 = base[(y1 << 6) + x1];
  const float val = v00 * (1.f - wx) * (1.f - wy) + v01 * wx * (1.f - wy)
                  + v10 * (1.f - wx) * wy         + v11 * wx * wy;
  rfb[(((a << 3) + n) * 64 + k) * 256 + c] = val;
}

// Stages 3/4: out[n,a,c,j] = b_j + sum_i softmax_i(a_i*b_j) * a_i
// Exact stable softmax: max_i(a_i*b_j) = max(b_j*max(a), b_j*min(a)).
// Block = (a, n, 4-channel slab); thread = (c_local, j). In-place-safe:
// all reads staged to LDS before any write.
__global__ __launch_bounds__(256) void rois_kernel(
    const float* __restrict__ f1, const float* __restrict__ f2,
    float* __restrict__ out)
{
  __shared__ float a_sh[4][65];   // pad -> conflict-free broadcast reads
  __shared__ float b_sh[4][65];
  __shared__ float amax[4], amin[4];

  const int bid = blockIdx.x;           // 9*8*64 = 4608
  const int aa  = bid / 512;
  const int rem = bid & 511;
  const int n   = rem >> 6;
  const int cbase = (rem & 63) << 2;

  const int tid = threadIdx.x;
  const int c_l = tid & 3;
  const int i   = tid >> 2;             // doubles as j in compute phase

  const int idx = (((aa << 3) + n) * 64 + i) * 256 + cbase + c_l;
  a_sh[c_l][i] = f1[idx];
  b_sh[c_l][i] = f2[idx];
  __syncthreads();

  if (tid < 4) {
    float mx = a_sh[tid][0], mn = mx;
    for (int q = 1; q < 64; ++q) {
      const float v = a_sh[tid][q];
      mx = fmaxf(mx, v); mn = fminf(mn, v);
    }
    amax[tid] = mx; amin[tid] = mn;
  }
  __syncthreads();

  const float bj = b_sh[c_l][i];
  const float mx = fmaxf(bj * amax[c_l], bj * amin[c_l]);
  float s = 0.f, t = 0.f;
  #pragma unroll 8
  for (int q = 0; q < 64; ++q) {
    const float ai = a_sh[c_l][q];
    const float e  = __expf(ai * bj - mx);
    s += e;
    t = fmaf(e, ai, t);
  }
  out[idx] = bj + t / s;
}

// ---------------------------------------------------------------------------
extern "C" void kernel_launch(void* const* d_in, const int* in_sizes, int n_in,
                              void* d_out, int out_size, void* d_ws, size_t ws_size,
                              hipStream_t stream) {
  (void)in_sizes; (void)n_in; (void)out_size; (void)ws_size;
  const float* rois_a = (const float*)d_in[1];
  const float* fb     = (const float*)d_in[2];
  const float* rois_c = (const float*)d_in[3];
  const float* wreg   = (const float*)d_in[4];
  const float* breg   = (const float*)d_in[5];
  float* out = (float*)d_out;

  // Workspace carve-up (~6.75 MB): every element written each call, no init.
  float* part = (float*)d_ws;              // 64*8*4*256   = 524288 f
  float* offs = part + 524288;             // 8*64*16      =   8192 f
  float* rfb  = offs + 8192;               // 4608*256     = 1179648 f

  conv_wmma_kernel <<<2048, 32, 0, stream>>>(fb, wreg, part);
  reduce_reg_kernel<<<  32, 256, 0, stream>>>(part, breg, offs);
  sample_kernel    <<<4608, 256, 0, stream>>>(fb, offs, rfb);
  rois_kernel      <<<4608, 256, 0, stream>>>(rois_a, rfb, rfb);   // in-place
  rois_kernel      <<<4608, 256, 0, stream>>>(rfb, rois_c, out);
}